// MultiLevelDDI_44865228374375
// MI455X (gfx1250) — compile-verified
//
#include <hip/hip_runtime.h>
#include <math.h>

// ---------------------------------------------------------------------------
// ProbSparse (Informer) attention for MI455X (gfx1250, wave32, WMMA + TDM).
// B=8, L=2048, H=8, D=64, u = U_part = 40.
// Memory-bound (~134MB @ 23.3TB/s). GEMMs: V_WMMA_F32_16X16X4_F32.
// K/V tile staging: tensor_load_to_lds (TDM) double-buffered on TENSORcnt.
// Row reductions: ds_swizzle_b32 immediate xor patterns (1 op per step).
// ---------------------------------------------------------------------------

typedef __attribute__((ext_vector_type(2))) float    v2f;
typedef __attribute__((ext_vector_type(8))) float    v8f;
typedef __attribute__((ext_vector_type(4))) unsigned u32x4;
typedef __attribute__((ext_vector_type(8))) int      i32x8;
typedef __attribute__((ext_vector_type(4))) int      i32x4;

#define B_  8
#define L_  2048
#define H_  8
#define D_  64
#define NS_ 40    // U_part == u == 40
#define UPAD_ 48  // 40 padded to 3 tiles of 16

// deterministic stand-in for jax.random.randint sample indices (l, s) -> [0,2048)
__device__ __forceinline__ unsigned sample_idx(unsigned l, unsigned s) {
    unsigned x = l * 2654435761u ^ (s * 0x9E3779B9u + 0x85EBCA6Bu);
    x ^= x >> 16; x *= 0x7FEB352Du;
    x ^= x >> 15; x *= 0x846CA68Bu;
    x ^= x >> 16;
    return x & (L_ - 1u);
}

// ds_swizzle_b32 group-of-32 xor pattern: new_lane = (lane & 0x1F) ^ XOR
template <int XOR>
__device__ __forceinline__ float swz_xor(float v) {
    return __int_as_float(
        __builtin_amdgcn_ds_swizzle(__float_as_int(v), 0x1F | (XOR << 10)));
}
// reduce over the 16 lanes of each half-wave (n-dimension of a WMMA C row)
__device__ __forceinline__ float red16_max(float v) {
    v = fmaxf(v, swz_xor<1>(v));
    v = fmaxf(v, swz_xor<2>(v));
    v = fmaxf(v, swz_xor<4>(v));
    v = fmaxf(v, swz_xor<8>(v));
    return v;
}
__device__ __forceinline__ float red16_sum(float v) {
    v += swz_xor<1>(v);
    v += swz_xor<2>(v);
    v += swz_xor<4>(v);
    v += swz_xor<8>(v);
    return v;
}

// generic pointer (LDS aperture) -> 32-bit LDS byte offset (flat addr[31:0])
__device__ __forceinline__ unsigned lds_byte_off(const void* p) {
    return (unsigned)(unsigned long long)(uintptr_t)p;
}

// ---------------------------------------------------------------------------
// TDM: DMA one 16-row x 64-float tile (row stride H*D floats) into LDS.
// D# per CDNA5 ISA ch.8: group0 = {count|..., lds_addr, gaddr lo, gaddr hi|type},
// group1 = {data_size, tensor dims, tile dims, dim0 stride}. 2D => groups 2/3 zero.
// This toolchain's builtin takes 6 operands (clang-23 form).
// ---------------------------------------------------------------------------
__device__ __forceinline__ void tdm_load_tile_16x64(const float* g, unsigned lds_off) {
    const unsigned long long ga = (unsigned long long)(uintptr_t)g;
    u32x4 g0;
    g0[0] = 1u;                                   // count=1, user descriptor
    g0[1] = lds_off;                              // lds_addr (bytes)
    g0[2] = (unsigned)(ga & 0xFFFFFFFFu);         // global_addr[31:0]
    g0[3] = ((unsigned)(ga >> 32) & 0x01FFFFFFu)  // global_addr[56:32]
            | 0x80000000u;                        // type=2 ("image") in [127:126]
    i32x8 g1;
    g1[0] = (int)(2u << 16);                      // data_size=2 -> 4-byte elements
    g1[1] = (int)(64u << 16);                     // tensor_dim0[15:0] = 64 (bits 79:48)
    g1[2] = (int)(2048u << 16);                   // tensor_dim1[15:0] = 2048 (bits 111:80)
    g1[3] = (int)(64u << 16);                     // tile_dim0 = 64 (bits 127:112)
    g1[4] = 16;                                   // tile_dim1 = 16, tile_dim2 = 0
    g1[5] = (int)(H_ * D_);                       // tensor_dim0_stride = 512 (bits 207:160)
    g1[6] = 0;
    g1[7] = 0;
    const i32x4 z4 = (i32x4){0, 0, 0, 0};
    const i32x8 z8 = (i32x8){0, 0, 0, 0, 0, 0, 0, 0};
    __builtin_amdgcn_tensor_load_to_lds(g0, g1, z4, z4, z8, 0);
}

// ---------------------------------------------------------------------------
// Kernel 1: mean over L of V  ->  mean[b,h,d]   (one block per (b,h))
// V layout: (B, L, H, D) => idx = ((b*L + l)*H + h)*D + d
// ---------------------------------------------------------------------------
__global__ void k_vmean(const float* __restrict__ V, float* __restrict__ mean) {
    __shared__ float sm[256];
    const int bh = blockIdx.x;          // 0..63
    const int b = bh >> 3, h = bh & 7;
    const int tid = threadIdx.x;        // 256
    const int d = tid & 63;
    const int c = tid >> 6;             // 4 chunks of 512 l's
    float acc = 0.f;
    const int lbeg = c * (L_ / 4), lend = lbeg + (L_ / 4);
    for (int l = lbeg; l < lend; ++l)
        acc += V[((size_t)(b * L_ + l) * H_ + h) * D_ + d];
    sm[tid] = acc;
    __syncthreads();
    if (c == 0) {
        float s = sm[d] + sm[64 + d] + sm[128 + d] + sm[192 + d];
        mean[(b * H_ + h) * D_ + d] = s * (1.0f / (float)L_);
    }
}

// ---------------------------------------------------------------------------
// Kernel 2: broadcast mean into out (B,L,H,D), coalesced float4
// ---------------------------------------------------------------------------
__global__ void k_bcast(const float* __restrict__ mean, float* __restrict__ out) {
    const size_t n4 = (size_t)B_ * L_ * H_ * (D_ / 4);
    const float4* m4 = (const float4*)mean;
    float4* o4 = (float4*)out;
    for (size_t i = (size_t)blockIdx.x * blockDim.x + threadIdx.x; i < n4;
         i += (size_t)gridDim.x * blockDim.x) {
        const int d4 = (int)(i & 15u);
        const int h  = (int)((i >> 4) & 7u);
        const int b  = (int)(i >> 18);
        o4[i] = m4[(b * H_ + h) * 16 + d4];
    }
}

// ---------------------------------------------------------------------------
// Kernel 3: sparsity measure  M[b,h,l] = max_s(q·k_s) - sum_s(q·k_s)/L
// One thread per (b,h,l).  K is L2-resident (4MB/batch << 192MB).
// ---------------------------------------------------------------------------
__global__ void k_measure(const float* __restrict__ Q, const float* __restrict__ K,
                          float* __restrict__ M) {
    const int gid = blockIdx.x * blockDim.x + threadIdx.x;  // B*H*L threads
    if (gid >= B_ * H_ * L_) return;
    const int l = gid & (L_ - 1);
    const int h = (gid >> 11) & 7;
    const int b = gid >> 14;
    const float4* qp = (const float4*)(Q + ((size_t)(b * L_ + l) * H_ + h) * D_);
    float4 q[16];
#pragma unroll
    for (int j = 0; j < 16; ++j) q[j] = qp[j];
    float mx = -INFINITY, sm = 0.f;
    for (int s = 0; s < NS_; ++s) {
        const unsigned key = sample_idx((unsigned)l, (unsigned)s);
        const float4* kp = (const float4*)(K + ((size_t)(b * L_ + key) * H_ + h) * D_);
        float dot = 0.f;
#pragma unroll
        for (int j = 0; j < 16; ++j) {
            float4 kv = kp[j];
            dot += q[j].x * kv.x + q[j].y * kv.y + q[j].z * kv.z + q[j].w * kv.w;
        }
        mx = fmaxf(mx, dot);
        sm += dot;
    }
    M[(b * H_ + h) * L_ + l] = mx - sm * (1.0f / (float)L_);
}

// ---------------------------------------------------------------------------
// Kernel 4: top-40 indices of M[b,h,:]  (one block per (b,h))
// ---------------------------------------------------------------------------
__global__ void k_topk(const float* __restrict__ M, int* __restrict__ Mtop) {
    __shared__ float vals[L_];
    __shared__ float rv[256];
    __shared__ int   ri[256];
    const int bh = blockIdx.x;
    const int tid = threadIdx.x;
    for (int i = tid; i < L_; i += 256) vals[i] = M[(size_t)bh * L_ + i];
    __syncthreads();
    for (int k = 0; k < NS_; ++k) {
        float best = -INFINITY; int bi = 0;
        for (int i = tid; i < L_; i += 256) {
            float v = vals[i];
            if (v > best) { best = v; bi = i; }
        }
        rv[tid] = best; ri[tid] = bi;
        __syncthreads();
        for (int s = 128; s > 0; s >>= 1) {
            if (tid < s && rv[tid + s] > rv[tid]) { rv[tid] = rv[tid + s]; ri[tid] = ri[tid + s]; }
            __syncthreads();
        }
        if (tid == 0) {
            Mtop[bh * UPAD_ + k] = ri[0];
            vals[ri[0]] = -INFINITY;
        }
        __syncthreads();
    }
}

// ---------------------------------------------------------------------------
// Kernel 5: flash attention for the 40 selected queries.
// grid = B*H*3, block = 32 (one wave owns one 16-row u-tile, streams all keys)
// S = Qr*K^T  (16 wmma / key-tile), O += P*V (16 wmma / key-tile).
// K/V tiles double-buffered in LDS via TDM (tensor_load_to_lds + TENSORcnt).
// ---------------------------------------------------------------------------
__global__ void k_flash(const float* __restrict__ Q, const float* __restrict__ K,
                        const float* __restrict__ V, const int* __restrict__ Mtop,
                        float* __restrict__ out) {
    __shared__ float Qt[16][64];
    __shared__ float Kt[2][16][64];   // TDM double buffer
    __shared__ float Vt[2][16][64];   // TDM double buffer
    __shared__ float Pt[16][17];      // +1 pad vs bank conflicts

    const int bh = blockIdx.x / 3;
    const int t  = blockIdx.x % 3;      // u-tile 0..2 (rows t*16 .. t*16+15)
    const int b = bh >> 3, h = bh & 7;
    const int lane = threadIdx.x;       // 0..31, wave32
    const int m16  = lane & 15;
    const int kk0  = (lane >> 4) * 2;   // 0 for lanes 0-15, 2 for lanes 16-31

    // ---- stage Q_reduce tile (gather by Mtop) into LDS; zero-pad rows >= 40
    {
        const int r = lane >> 1, half = lane & 1;
        const int u = t * 16 + r;
        if (u < NS_) {
            const int lg = Mtop[bh * UPAD_ + u];
            const float4* qp =
                (const float4*)(Q + ((size_t)(b * L_ + lg) * H_ + h) * D_ + half * 32);
            float4* dst = (float4*)(&Qt[r][half * 32]);
#pragma unroll
            for (int j = 0; j < 8; ++j) dst[j] = qp[j];
        } else {
            float4* dst = (float4*)(&Qt[r][half * 32]);
            const float4 z = make_float4(0.f, 0.f, 0.f, 0.f);
#pragma unroll
            for (int j = 0; j < 8; ++j) dst[j] = z;
        }
    }
    __syncthreads();

    // ---- resident A fragments of Qr: 16 chunks of K=4 (f32 A: 2 VGPRs each)
    v2f a[16];
#pragma unroll
    for (int c = 0; c < 16; ++c) {
        a[c].x = Qt[m16][4 * c + kk0];
        a[c].y = Qt[m16][4 * c + kk0 + 1];
    }

    // ---- flash state: rows r (lanes 0-15) / r+8 (lanes 16-31)
    float mrow[8], lrow[8];
    v8f   O[4];
#pragma unroll
    for (int r = 0; r < 8; ++r) { mrow[r] = -INFINITY; lrow[r] = 0.f; }
#pragma unroll
    for (int dt = 0; dt < 4; ++dt) O[dt] = (v8f){};

    // ---- prime the TDM pipeline with tile 0 (2 DMAs: K tile + V tile)
    {
        const size_t gb0 = ((size_t)(b * L_) * H_ + h) * D_;
        tdm_load_tile_16x64(K + gb0, lds_byte_off(&Kt[0][0][0]));
        tdm_load_tile_16x64(V + gb0, lds_byte_off(&Vt[0][0][0]));
    }

    for (int nt = 0; nt < L_ / 16; ++nt) {
        const int nb = nt & 1;
        // issue next tile's DMAs into the other buffer, then wait for this tile
        if (nt + 1 < L_ / 16) {
            const size_t gbn = ((size_t)(b * L_ + (nt + 1) * 16) * H_ + h) * D_;
            tdm_load_tile_16x64(K + gbn, lds_byte_off(&Kt[nb ^ 1][0][0]));
            tdm_load_tile_16x64(V + gbn, lds_byte_off(&Vt[nb ^ 1][0][0]));
            __builtin_amdgcn_s_wait_tensorcnt(2);   // tile nt's 2 DMAs complete
        } else {
            __builtin_amdgcn_s_wait_tensorcnt(0);
        }
        asm volatile("" ::: "memory");   // keep LDS reads below the wait

        // S = Qr(16x64) * K^T(64x16)  via 16 x WMMA_F32_16X16X4_F32
        v8f S = (v8f){};
#pragma unroll
        for (int c = 0; c < 16; ++c) {
            v2f bb;
            bb.x = Kt[nb][m16][4 * c + kk0];        // B[k][n] = K[key n][4c+k]
            bb.y = Kt[nb][m16][4 * c + kk0 + 1];
            S = __builtin_amdgcn_wmma_f32_16x16x4_f32(
                    false, a[c], false, bb, (short)0, S, false, false);
        }

        // flash update: scale 1/sqrt(64); row stats via ds_swizzle xor reductions
#pragma unroll
        for (int r = 0; r < 8; ++r) {
            float s = S[r] * 0.125f;
            const float red = red16_max(s);
            const float mn = fmaxf(mrow[r], red);
            const float corr = __expf(mrow[r] - mn);
            mrow[r] = mn;
            const float p = __expf(s - mn);
            const float rs = red16_sum(p);
            lrow[r] = lrow[r] * corr + rs;
#pragma unroll
            for (int dt = 0; dt < 4; ++dt) O[dt][r] *= corr;
            Pt[(lane < 16) ? r : (r + 8)][m16] = p;   // C-layout -> LDS
        }
        __syncthreads();

        // re-read P in A-layout, O += P(16x16) * V(16x64): 4x4 WMMAs
        v2f pa[4];
#pragma unroll
        for (int c = 0; c < 4; ++c) {
            pa[c].x = Pt[m16][4 * c + kk0];
            pa[c].y = Pt[m16][4 * c + kk0 + 1];
        }
#pragma unroll
        for (int dt = 0; dt < 4; ++dt) {
#pragma unroll
            for (int c = 0; c < 4; ++c) {
                v2f bv;
                bv.x = Vt[nb][4 * c + kk0][dt * 16 + m16];  // B[k][n] = V[4c+k][dt*16+n]
                bv.y = Vt[nb][4 * c + kk0 + 1][dt * 16 + m16];
                O[dt] = __builtin_amdgcn_wmma_f32_16x16x4_f32(
                            false, pa[c], false, bv, (short)0, O[dt], false, false);
            }
        }
        __syncthreads();   // Pt reuse next iteration
    }

    // ---- normalize by 1/l and scatter the 40 rows over the broadcast mean
#pragma unroll
    for (int r = 0; r < 8; ++r) {
        const int row = (lane < 16) ? r : (r + 8);
        const int u = t * 16 + row;
        const float inv = (lrow[r] > 0.f) ? (1.0f / lrow[r]) : 0.f;
        if (u < NS_) {
            const int lg = Mtop[bh * UPAD_ + u];
            float* ob = out + ((size_t)(b * L_ + lg) * H_ + h) * D_;
#pragma unroll
            for (int dt = 0; dt < 4; ++dt)
                ob[dt * 16 + m16] = O[dt][r] * inv;
        }
    }
}

// ---------------------------------------------------------------------------
extern "C" void kernel_launch(void* const* d_in, const int* in_sizes, int n_in,
                              void* d_out, int out_size, void* d_ws, size_t ws_size,
                              hipStream_t stream) {
    const float* Q = (const float*)d_in[0];
    const float* K = (const float*)d_in[1];
    const float* V = (const float*)d_in[2];
    float* out = (float*)d_out;

    // workspace layout
    float* mean = (float*)d_ws;                          // B*H*D
    float* M    = mean + B_ * H_ * D_;                   // B*H*L
    int*   Mtop = (int*)(M + (size_t)B_ * H_ * L_);      // B*H*UPAD

    k_vmean  <<<B_ * H_,                 256, 0, stream>>>(V, mean);
    k_bcast  <<<8192,                    256, 0, stream>>>(mean, out);
    k_measure<<<(B_ * H_ * L_) / 256,    256, 0, stream>>>(Q, K, M);
    k_topk   <<<B_ * H_,                 256, 0, stream>>>(M, Mtop);
    k_flash  <<<B_ * H_ * 3,              32, 0, stream>>>(Q, K, V, Mtop, out);
}